// SAGE_Linear_23081154248744
// MI455X (gfx1250) — compile-verified
//
#include <hip/hip_runtime.h>
#include <math.h>

// ---------------------------------------------------------------------------
// GraphSAGE link prediction on MI455X (gfx1250, wave32).
// ~24 GFLOP fp32; all feature tensors (~20MB each) are L2-resident (192MB L2),
// so the pipeline is gather/scatter + L2-traffic bound. Keep exact fp32 via
// V_WMMA_F32_16X16X4_F32. Each block computes 32 rows x 256 cols; each wave
// owns a 2x2 grid of 16x16 tiles so every weight fragment feeds 2 WMMAs
// (halves L2 weight re-fetch vs a 16-row block).
// ---------------------------------------------------------------------------

#define N_NODES 20000
#define N_EDGES 320000
#define N_QUERY 100000
#define D       256

typedef __attribute__((ext_vector_type(2))) float v2f;
typedef __attribute__((ext_vector_type(8))) float v8f;

static __device__ __forceinline__ v8f wmma_f32_k4(v2f a, v2f b, v8f c) {
  // 8 args: (neg_a, A, neg_b, B, c_mod, C, reuse_a, reuse_b)
  return __builtin_amdgcn_wmma_f32_16x16x4_f32(false, a, false, b, (short)0, c,
                                               false, false);
}

// ------------------------------- utility -----------------------------------

__global__ void zero_f32(float* __restrict__ p, int n) {
  int i = blockIdx.x * blockDim.x + threadIdx.x;
  if (i < n) p[i] = 0.0f;
}

__global__ void deg_scatter(const int* __restrict__ row, float* __restrict__ deg) {
  int e = blockIdx.x * blockDim.x + threadIdx.x;
  if (e < N_EDGES) atomicAdd(&deg[row[e]], 1.0f);
}

// agg[row[e], :] += x[col[e], :]   (one thread per edge x 4-channel group)
__global__ void msg_scatter(const int* __restrict__ row, const int* __restrict__ col,
                            const float* __restrict__ x, float* __restrict__ agg) {
  long long t = (long long)blockIdx.x * blockDim.x + threadIdx.x;
  int e = (int)(t >> 6);          // 64 groups of 4 channels per edge
  if (e >= N_EDGES) return;
  int cg = ((int)t & 63) << 2;
  int r = row[e], c = col[e];
  const float4 v = *(const float4*)(x + (size_t)c * D + cg);
  float* dst = agg + (size_t)r * D + cg;
  atomicAdd(dst + 0, v.x);
  atomicAdd(dst + 1, v.y);
  atomicAdd(dst + 2, v.z);
  atomicAdd(dst + 3, v.w);
}

// ----------------------- fused dual GEMM (SAGE layer) ----------------------
// out[m,n] = act( (agg[m,:]/max(deg[m],1)) @ Wl + x[m,:] @ Wr + bias[n] )
// grid.x = rows/32 ; block = 256 threads = 8 wave32. Wave w owns column tiles
// n0=32w and n0+16 and row tiles 0 and 16 of the 32-row block (2x2 tiles).
template <bool RELU>
__global__ __launch_bounds__(256)
void sage_gemm(const float* __restrict__ agg, const float* __restrict__ deg,
               const float* __restrict__ x,
               const float* __restrict__ Wl, const float* __restrict__ Wr,
               const float* __restrict__ bias, float* __restrict__ out) {
  __shared__ float sA[32][D + 4];
  __shared__ float sX[32][D + 4];
  const int tid  = threadIdx.x;
  const int row0 = blockIdx.x * 32;

  // Stage 32 rows of (mean-scaled agg) and x into LDS.
  for (int i = tid; i < 32 * (D / 4); i += 256) {
    int m  = i >> 6;            // row within tile (64 float4 groups per row)
    int c4 = (i & 63) << 2;
    int row = row0 + m;
    float invd = 1.0f / fmaxf(deg[row], 1.0f);
    float4 a = *(const float4*)(agg + (size_t)row * D + c4);
    sA[m][c4 + 0] = a.x * invd;
    sA[m][c4 + 1] = a.y * invd;
    sA[m][c4 + 2] = a.z * invd;
    sA[m][c4 + 3] = a.w * invd;
    float4 b = *(const float4*)(x + (size_t)row * D + c4);
    sX[m][c4 + 0] = b.x;
    sX[m][c4 + 1] = b.y;
    sX[m][c4 + 2] = b.z;
    sX[m][c4 + 3] = b.w;
  }
  __syncthreads();

  const int wave = tid >> 5;
  const int lane = tid & 31;
  const int m    = lane & 15;
  const int koff = (lane < 16) ? 0 : 2;   // A/B K sub-row per half-wave
  const int n0   = wave * 32;
  const int nA   = n0 + m;
  const int nB   = n0 + 16 + m;

  v8f acc00 = {};   // rows [0,16)  x cols [n0,n0+16)
  v8f acc01 = {};   // rows [0,16)  x cols [n0+16,n0+32)
  v8f acc10 = {};   // rows [16,32) x cols [n0,n0+16)
  v8f acc11 = {};   // rows [16,32) x cols [n0+16,n0+32)

  for (int kb = 0; kb < D; kb += 4) {
    const int k = kb + koff;
    v2f aA0; aA0[0] = sA[m][k];      aA0[1] = sA[m][k + 1];
    v2f aA1; aA1[0] = sA[m + 16][k]; aA1[1] = sA[m + 16][k + 1];
    v2f aX0; aX0[0] = sX[m][k];      aX0[1] = sX[m][k + 1];
    v2f aX1; aX1[0] = sX[m + 16][k]; aX1[1] = sX[m + 16][k + 1];
    v2f bl0; bl0[0] = Wl[(size_t)k * D + nA]; bl0[1] = Wl[(size_t)(k + 1) * D + nA];
    v2f bl1; bl1[0] = Wl[(size_t)k * D + nB]; bl1[1] = Wl[(size_t)(k + 1) * D + nB];
    v2f br0; br0[0] = Wr[(size_t)k * D + nA]; br0[1] = Wr[(size_t)(k + 1) * D + nA];
    v2f br1; br1[0] = Wr[(size_t)k * D + nB]; br1[1] = Wr[(size_t)(k + 1) * D + nB];
    acc00 = wmma_f32_k4(aA0, bl0, acc00);
    acc00 = wmma_f32_k4(aX0, br0, acc00);
    acc01 = wmma_f32_k4(aA0, bl1, acc01);
    acc01 = wmma_f32_k4(aX0, br1, acc01);
    acc10 = wmma_f32_k4(aA1, bl0, acc10);
    acc10 = wmma_f32_k4(aX1, br0, acc10);
    acc11 = wmma_f32_k4(aA1, bl1, acc11);
    acc11 = wmma_f32_k4(aX1, br1, acc11);
  }

  // Epilogue: C/D layout -> VGPR r holds row r (lanes 0-15) / r+8 (lanes 16-31)
  const int roff = (lane < 16) ? 0 : 8;
  const float bvA = bias[nA];
  const float bvB = bias[nB];
#pragma unroll
  for (int r = 0; r < 8; ++r) {
    float v00 = acc00[r] + bvA;
    float v01 = acc01[r] + bvB;
    float v10 = acc10[r] + bvA;
    float v11 = acc11[r] + bvB;
    if (RELU) {
      v00 = fmaxf(v00, 0.0f); v01 = fmaxf(v01, 0.0f);
      v10 = fmaxf(v10, 0.0f); v11 = fmaxf(v11, 0.0f);
    }
    out[(size_t)(row0 + r + roff) * D + nA]      = v00;
    out[(size_t)(row0 + r + roff) * D + nB]      = v01;
    out[(size_t)(row0 + 16 + r + roff) * D + nA] = v10;
    out[(size_t)(row0 + 16 + r + roff) * D + nB] = v11;
  }
}

// --------------------------- fused link predictor --------------------------
// out[q] = sigmoid( relu((x2[e0[q]]*x2[e1[q]]) @ Wp1 + bp1) @ Wp2 + bp2 )
// 32 queries per block; waves own 2x2 tile grids like sage_gemm.
__global__ __launch_bounds__(256)
void predictor(const int* __restrict__ e0, const int* __restrict__ e1,
               const float* __restrict__ x2,
               const float* __restrict__ Wp1, const float* __restrict__ bp1,
               const float* __restrict__ Wp2, const float* __restrict__ bp2,
               float* __restrict__ out) {
  __shared__ float sH[32][D + 4];
  __shared__ float sP[32][9];
  const int tid = threadIdx.x;
  const int q0  = blockIdx.x * 32;

  // Stage h_in = x2[e0] * x2[e1] for 32 queries.
  for (int i = tid; i < 32 * (D / 4); i += 256) {
    int m  = i >> 6;
    int c4 = (i & 63) << 2;
    int q  = q0 + m;
    int s = e0[q], d = e1[q];
    float4 a = *(const float4*)(x2 + (size_t)s * D + c4);
    float4 b = *(const float4*)(x2 + (size_t)d * D + c4);
    sH[m][c4 + 0] = a.x * b.x;
    sH[m][c4 + 1] = a.y * b.y;
    sH[m][c4 + 2] = a.z * b.z;
    sH[m][c4 + 3] = a.w * b.w;
  }
  __syncthreads();

  const int wave = tid >> 5;
  const int lane = tid & 31;
  const int m    = lane & 15;
  const int koff = (lane < 16) ? 0 : 2;
  const int n0   = wave * 32;
  const int nA   = n0 + m;
  const int nB   = n0 + 16 + m;

  v8f acc00 = {};
  v8f acc01 = {};
  v8f acc10 = {};
  v8f acc11 = {};
  for (int kb = 0; kb < D; kb += 4) {
    const int k = kb + koff;
    v2f a0; a0[0] = sH[m][k];      a0[1] = sH[m][k + 1];
    v2f a1; a1[0] = sH[m + 16][k]; a1[1] = sH[m + 16][k + 1];
    v2f b0; b0[0] = Wp1[(size_t)k * D + nA]; b0[1] = Wp1[(size_t)(k + 1) * D + nA];
    v2f b1; b1[0] = Wp1[(size_t)k * D + nB]; b1[1] = Wp1[(size_t)(k + 1) * D + nB];
    acc00 = wmma_f32_k4(a0, b0, acc00);
    acc01 = wmma_f32_k4(a0, b1, acc01);
    acc10 = wmma_f32_k4(a1, b0, acc10);
    acc11 = wmma_f32_k4(a1, b1, acc11);
  }
  __syncthreads();   // everyone done reading sH before it is overwritten

  const int roff = (lane < 16) ? 0 : 8;
  const float bvA = bp1[nA];
  const float bvB = bp1[nB];
#pragma unroll
  for (int r = 0; r < 8; ++r) {
    sH[r + roff][nA]      = fmaxf(acc00[r] + bvA, 0.0f);
    sH[r + roff][nB]      = fmaxf(acc01[r] + bvB, 0.0f);
    sH[16 + r + roff][nA] = fmaxf(acc10[r] + bvA, 0.0f);
    sH[16 + r + roff][nB] = fmaxf(acc11[r] + bvB, 0.0f);
  }
  __syncthreads();

  // Reduce against Wp2: 32 rows x 8 partials (32 cols each), then final sum.
  {
    int row  = tid >> 3;          // 0..31
    int part = tid & 7;           // 0..7
    float s = 0.0f;
#pragma unroll
    for (int j = 0; j < 32; ++j) {
      int c = part * 32 + j;
      s += sH[row][c] * Wp2[c];
    }
    sP[row][part] = s;
  }
  __syncthreads();
  if (tid < 32) {
    float s = bp2[0];
#pragma unroll
    for (int j = 0; j < 8; ++j) s += sP[tid][j];
    out[q0 + tid] = 1.0f / (1.0f + __expf(-s));
  }
}

// ------------------------------ launcher -----------------------------------

extern "C" void kernel_launch(void* const* d_in, const int* in_sizes, int n_in,
                              void* d_out, int out_size, void* d_ws, size_t ws_size,
                              hipStream_t stream) {
  const int*   adj_row = (const int*)d_in[0];
  const int*   adj_col = (const int*)d_in[1];
  const int*   edges   = (const int*)d_in[2];   // [2, Q] row-major
  const float* emb     = (const float*)d_in[3];
  const float* W1l     = (const float*)d_in[4];
  const float* b1      = (const float*)d_in[5];
  const float* W1r     = (const float*)d_in[6];
  const float* W2l     = (const float*)d_in[7];
  const float* b2      = (const float*)d_in[8];
  const float* W2r     = (const float*)d_in[9];
  const float* Wp1     = (const float*)d_in[10];
  const float* bp1     = (const float*)d_in[11];
  const float* Wp2     = (const float*)d_in[12];
  const float* bp2     = (const float*)d_in[13];
  float*       out     = (float*)d_out;

  // Workspace carve-up (all 16B aligned): deg | agg | x1 | x2  (~61.5 MB)
  char* ws = (char*)d_ws;
  float* deg = (float*)ws;  ws += (size_t)N_NODES * sizeof(float);        // 80000 B (16B-mult)
  float* agg = (float*)ws;  ws += (size_t)N_NODES * D * sizeof(float);
  float* x1  = (float*)ws;  ws += (size_t)N_NODES * D * sizeof(float);
  float* x2  = (float*)ws;  ws += (size_t)N_NODES * D * sizeof(float);

  const int ND = N_NODES * D;
  const long long scat_threads = (long long)N_EDGES * 64;
  const int scat_blocks = (int)((scat_threads + 255) / 256);

  // deg (shared by both layers) ------------------------------------------------
  zero_f32<<<(N_NODES + 255) / 256, 256, 0, stream>>>(deg, N_NODES);
  deg_scatter<<<(N_EDGES + 255) / 256, 256, 0, stream>>>(adj_row, deg);

  // Layer 1: agg = scatter(emb); x1 = relu(mean(agg)@W1l + b1 + emb@W1r) ------
  zero_f32<<<(ND + 255) / 256, 256, 0, stream>>>(agg, ND);
  msg_scatter<<<scat_blocks, 256, 0, stream>>>(adj_row, adj_col, emb, agg);
  sage_gemm<true><<<N_NODES / 32, 256, 0, stream>>>(agg, deg, emb, W1l, W1r, b1, x1);

  // Layer 2: agg = scatter(x1); x2 = mean(agg)@W2l + b2 + x1@W2r --------------
  zero_f32<<<(ND + 255) / 256, 256, 0, stream>>>(agg, ND);
  msg_scatter<<<scat_blocks, 256, 0, stream>>>(adj_row, adj_col, x1, agg);
  sage_gemm<false><<<N_NODES / 32, 256, 0, stream>>>(agg, deg, x1, W2l, W2r, b2, x2);

  // Link predictor ------------------------------------------------------------
  predictor<<<N_QUERY / 32, 256, 0, stream>>>(edges, edges + N_QUERY, x2,
                                              Wp1, bp1, Wp2, bp2, out);
}